// ROIAlign_84851373900423
// MI455X (gfx1250) — compile-verified
//
#include <hip/hip_runtime.h>
#include <hip/hip_bf16.h>

// ROI Align (R=1000, C=256, H=200, W=304, PH=PW=7, SR=2, SCALE=0.25) for gfx1250.
//
// Strategy: one wave32 owns (roi, 16-channel tile). Lanes gather bilinear
// samples (the real cost; feature map is L2-resident), and the per-bin 2x2
// averaging O[bin,ch] = sum_s Avg[bin,s] * S[s,ch] is done on the matrix pipe
// with V_WMMA_F32_16X16X4_F32 (full fp32, Avg entries are exactly 0.25).
// K = 196 samples in 49 slices of 4; M = bins padded 49->64 (4 acc tiles);
// N = 16 channels. Tile selection per K-slice is wave-uniform, so EXEC is
// all-1s at every WMMA as the ISA requires.

#define N_IMG 2
#define C_CH 256
#define H_F 200
#define W_F 304
#define PHB 7
#define PWB 7
#define NBIN 49      // 7*7
#define NSAMP 196    // 14*14

typedef float v2f __attribute__((ext_vector_type(2)));
typedef float v8f __attribute__((ext_vector_type(8)));

__device__ __forceinline__ int bin_of(int s) {
    int iy = s / 14;
    int ix = s - iy * 14;
    return (iy >> 1) * PWB + (ix >> 1);
}

// One bilinear sample of channel plane `cb` (H_F x W_F), sample index s in 14x14 grid.
// Matches the reference _axis_interp semantics exactly (clip, floor, frac, validity mask).
__device__ __forceinline__ float roi_sample(const float* __restrict__ cb, int s,
                                            float y1, float x1, float bh, float bw) {
    int iy = s / 14;
    int ix = s - iy * 14;
    float yc = y1 + ((float)iy + 0.5f) * 0.5f * bh;   // gy = (iy+0.5)/SR, SR=2
    float xc = x1 + ((float)ix + 0.5f) * 0.5f * bw;
    bool ok = (yc > -1.0f) && (yc < (float)H_F) && (xc > -1.0f) && (xc < (float)W_F);
    float y = fminf(fmaxf(yc, 0.0f), (float)(H_F - 1));
    float x = fminf(fmaxf(xc, 0.0f), (float)(W_F - 1));
    int y0 = (int)floorf(y); y0 = y0 > (H_F - 2) ? (H_F - 2) : y0;
    int x0 = (int)floorf(x); x0 = x0 > (W_F - 2) ? (W_F - 2) : x0;
    float fy = y - (float)y0;
    float fx = x - (float)x0;
    const float* p = cb + y0 * W_F + x0;
    float v00 = p[0], v01 = p[1], v10 = p[W_F], v11 = p[W_F + 1];
    float top = v00 + fx * (v01 - v00);
    float bot = v10 + fx * (v11 - v10);
    float val = top + fy * (bot - top);
    return ok ? val : 0.0f;
}

__global__ __launch_bounds__(128)
void roi_align_wmma_kernel(const float* __restrict__ feat,
                           const float* __restrict__ rois,
                           float* __restrict__ out) {
    const int lane = threadIdx.x & 31;
    const int w    = threadIdx.x >> 5;          // wave in block: 0..3
    const int r    = blockIdx.x >> 2;           // ROI index
    const int ct   = ((blockIdx.x & 3) << 2) + w; // channel tile 0..15 (16 ch each)

    // ---- per-ROI parameters (broadcast loads, uniform across the wave) ----
    const float* roi = rois + r * 5;
    const int   b  = (int)roi[0];
    const float x1 = roi[1] * 0.25f - 0.5f;
    const float y1 = roi[2] * 0.25f - 0.5f;
    const float x2 = roi[3] * 0.25f - 0.5f;
    const float y2 = roi[4] * 0.25f - 0.5f;
    const float bin_h = (y2 - y1) * (1.0f / (float)PHB);
    const float bin_w = (x2 - x1) * (1.0f / (float)PWB);

    // ---- WMMA lane roles (ISA 16x16x4 f32 layouts) ----
    const int m    = lane & 15;   // A row / B,N column / D column
    const int half = lane >> 4;   // selects K pair {2h, 2h+1}
    const int chan = ct * 16 + m;
    const float* cbase = feat + ((size_t)(b * C_CH + chan)) * (H_F * W_F);

    v8f acc0 = {}; v8f acc1 = {}; v8f acc2 = {}; v8f acc3 = {};

    for (int k = 0; k < 49; ++k) {
        const int sbase = 4 * k;
        const int s0 = sbase + half * 2;

        // B operand: this lane's 2 sample values for channel `chan`
        float sv0 = roi_sample(cbase, s0,     y1, x1, bin_h, bin_w);
        float sv1 = roi_sample(cbase, s0 + 1, y1, x1, bin_h, bin_w);
        v2f Bv = {sv0, sv1};

        const int binA = bin_of(s0);
        const int binB = bin_of(s0 + 1);

        // Which 16-bin M-tiles does this K-slice touch? (uniform: depends on k only)
        unsigned tmask = 0;
#pragma unroll
        for (int j = 0; j < 4; ++j) tmask |= 1u << (bin_of(sbase + j) >> 4);

        if (tmask & 1u) {
            v2f Av = { binA == m        ? 0.25f : 0.0f, binB == m        ? 0.25f : 0.0f };
            acc0 = __builtin_amdgcn_wmma_f32_16x16x4_f32(false, Av, false, Bv, (short)0, acc0, false, false);
        }
        if (tmask & 2u) {
            v2f Av = { binA == 16 + m   ? 0.25f : 0.0f, binB == 16 + m   ? 0.25f : 0.0f };
            acc1 = __builtin_amdgcn_wmma_f32_16x16x4_f32(false, Av, false, Bv, (short)0, acc1, false, false);
        }
        if (tmask & 4u) {
            v2f Av = { binA == 32 + m   ? 0.25f : 0.0f, binB == 32 + m   ? 0.25f : 0.0f };
            acc2 = __builtin_amdgcn_wmma_f32_16x16x4_f32(false, Av, false, Bv, (short)0, acc2, false, false);
        }
        if (tmask & 8u) {
            v2f Av = { binA == 48 + m   ? 0.25f : 0.0f, binB == 48 + m   ? 0.25f : 0.0f };
            acc3 = __builtin_amdgcn_wmma_f32_16x16x4_f32(false, Av, false, Bv, (short)0, acc3, false, false);
        }
    }

    // ---- stage D tiles to LDS as [ch][bin], then coalesced store ----
    // D layout: VGPR j holds row M = j + 8*half, column N = m.
    __shared__ float sOut[4][16 * NBIN];   // 4 waves * 784 floats = 12544 B
    float* my = &sOut[w][0];
#pragma unroll
    for (int j = 0; j < 8; ++j) {
        const int mr = j + 8 * half;             // 0..15
        my[m * NBIN + mr]      = acc0[j];        // bins  0..15
        my[m * NBIN + 16 + mr] = acc1[j];        // bins 16..31
        my[m * NBIN + 32 + mr] = acc2[j];        // bins 32..47
    }
    if (half == 0 && m < 16) {                   // bin 48 only (tile 3, row 0)
        my[m * NBIN + 48] = acc3[0];
    }
    // same-wave LDS producer/consumer: hardware keeps DS ops in order, no barrier.

    // Output (R,C,7,7): channels of this tile are contiguous -> 784 contiguous floats.
    float* op = out + ((size_t)r * C_CH + (size_t)ct * 16) * NBIN;
    for (int i = lane; i < 16 * NBIN; i += 32) op[i] = my[i];
}

extern "C" void kernel_launch(void* const* d_in, const int* in_sizes, int n_in,
                              void* d_out, int out_size, void* d_ws, size_t ws_size,
                              hipStream_t stream) {
    const float* feat = (const float*)d_in[0];   // (2,256,200,304) f32
    const float* rois = (const float*)d_in[1];   // (R,5) f32
    float* out = (float*)d_out;                  // (R,256,7,7) f32

    const int R = in_sizes[1] / 5;
    if (R <= 0) return;
    // 4 channel-tiles per block (4 waves), 16 tiles per ROI -> 4 blocks per ROI.
    roi_align_wmma_kernel<<<dim3(R * 4), dim3(128), 0, stream>>>(feat, rois, out);
}